// BlockLinear_63496796504583
// MI455X (gfx1250) — compile-verified
//
#include <hip/hip_runtime.h>

typedef float v2f __attribute__((ext_vector_type(2)));
typedef float v4f __attribute__((ext_vector_type(4)));
typedef float v8f __attribute__((ext_vector_type(8)));

namespace {
constexpr int kH       = 16;      // number of diagonal blocks
constexpr int kD       = 64;      // block size
constexpr int kB       = 32768;   // batch (columns)
constexpr int kThreads = 256;     // 8 waves (wave32)
constexpr int kWaves   = kThreads / 32;
constexpr int kNT      = 16;      // output columns per wave
constexpr int kNB      = kWaves * kNT;  // 128 columns per block
}

__global__ __launch_bounds__(kThreads)
void block_linear_wmma_f32(const float* __restrict__ W,
                           const float* __restrict__ X,
                           float* __restrict__ Y) {
  // One 64x64 fp32 weight block in LDS (16 KB of the 320 KB WGP pool).
  __shared__ float Wl[kD * kD];

  const int tid = threadIdx.x;
  const int h   = blockIdx.y;

  // Cooperative, fully coalesced b128 load of W_h into LDS.
  {
    const v4f* __restrict__ Wg4 = (const v4f*)(W + (size_t)h * kD * kD);
    v4f* Wl4 = (v4f*)Wl;
#pragma unroll
    for (int i = 0; i < (kD * kD / 4) / kThreads; ++i)
      Wl4[tid + i * kThreads] = Wg4[tid + i * kThreads];
  }
  __syncthreads();

  const int wave    = tid >> 5;
  const int lane    = tid & 31;
  const int lane_lo = lane & 15;   // N (B-frag / C tile) or M (A-frag) position
  const int khalf   = lane >> 4;   // 0 -> K pair {0,1}, 1 -> K pair {2,3}

  const int n0  = blockIdx.x * kNB + wave * kNT;
  const int col = n0 + lane_lo;

  const float* __restrict__ Xh = X + (size_t)h * kD * kB;
  float*       __restrict__ Yh = Y + (size_t)h * kD * kB;

  // Four 16x16 f32 accumulators cover M = 0..63 for this wave's 16 columns.
  v8f c0 = {}, c1 = {}, c2 = {}, c3 = {};

#pragma unroll
  for (int kk = 0; kk < kD / 4; ++kk) {
    const int krow = kk * 4 + khalf * 2;

    // B-fragment (4x16 f32): lane holds column `col`, K pair (krow, krow+1).
    // Each global_load_b32 covers two fully-used 64B row segments.
    v2f b;
    b.x = Xh[(size_t)krow * kB + col];
    b.y = Xh[(size_t)(krow + 1) * kB + col];

    // A-fragments (16x4 f32) for the 4 M-tiles: two consecutive f32 per lane
    // from the LDS copy of W -> single ds_load_b64 each (8B aligned).
    const v2f a0 = *(const v2f*)&Wl[(0 * 16 + lane_lo) * kD + krow];
    const v2f a1 = *(const v2f*)&Wl[(1 * 16 + lane_lo) * kD + krow];
    const v2f a2 = *(const v2f*)&Wl[(2 * 16 + lane_lo) * kD + krow];
    const v2f a3 = *(const v2f*)&Wl[(3 * 16 + lane_lo) * kD + krow];

    // D = A x B + C in fp32 (RNE), matching the fp32 reference einsum.
    c0 = __builtin_amdgcn_wmma_f32_16x16x4_f32(false, a0, false, b, (short)0, c0, false, false);
    c1 = __builtin_amdgcn_wmma_f32_16x16x4_f32(false, a1, false, b, (short)0, c1, false, false);
    c2 = __builtin_amdgcn_wmma_f32_16x16x4_f32(false, a2, false, b, (short)0, c2, false, false);
    c3 = __builtin_amdgcn_wmma_f32_16x16x4_f32(false, a3, false, b, (short)0, c3, false, false);
  }

  // C/D layout (32-bit, 16x16): VGPR v -> row (v + 8*khalf), N = lane_lo.
  // Each global_store_b32 writes two fully-used 64B row segments.
#pragma unroll
  for (int v = 0; v < 8; ++v) {
    const int rbase = v + khalf * 8;
    Yh[(size_t)(0 * 16 + rbase) * kB + col] = c0[v];
    Yh[(size_t)(1 * 16 + rbase) * kB + col] = c1[v];
    Yh[(size_t)(2 * 16 + rbase) * kB + col] = c2[v];
    Yh[(size_t)(3 * 16 + rbase) * kB + col] = c3[v];
  }
}

extern "C" void kernel_launch(void* const* d_in, const int* in_sizes, int n_in,
                              void* d_out, int out_size, void* d_ws, size_t ws_size,
                              hipStream_t stream) {
  const float* W = (const float*)d_in[0];  // [16, 64, 64] fp32
  const float* X = (const float*)d_in[1];  // [1024, 32768] fp32
  float*       Y = (float*)d_out;          // [1024, 32768] fp32

  dim3 grid(kB / kNB, kH);  // (256, 16) blocks
  block_linear_wmma_f32<<<grid, kThreads, 0, stream>>>(W, X, Y);
}